// ShiftNMF_14783277433385
// MI455X (gfx1250) — compile-verified
//
#include <hip/hip_runtime.h>
#include <math.h>

// ShiftNMF forward for MI455X (gfx1250, wave32, WMMA).
//
//   Hf = FFT_4096( softplus(H) )            (32 rows)  -> four-step FFT, 64-pt
//        DFTs done as 64x64x64 matmuls on V_WMMA_F32_16X16X4_F32
//   V[n,m] = sum_d softplus(W)[n,d] * exp(-2pi i tau[n,d] m/M) * Hf[d,m]
//        -> incremental complex rotation (2 sincos per (lane,d) instead of 8)

#define MDIM 4096
#define NDIM 1024
#define RANKD 32
#define PI_F 3.14159265358979323846f

typedef float v2f __attribute__((ext_vector_type(2)));
typedef float v8f __attribute__((ext_vector_type(8)));

__device__ __forceinline__ float softplus_f(float x) {
    return (x > 20.0f) ? x : log1pf(__expf(x));
}

// ---------------------------------------------------------------------------
// Kernel 1: per-rank-row 4096-pt FFT of softplus(H), four-step (64 x 64),
// 64-pt DFTs as WMMA f32 16x16x4 matmuls. One workgroup (256 thr = 8 waves)
// per rank row d.  Hf written interleaved complex (re,im) to d_ws.
// ---------------------------------------------------------------------------
__global__ __launch_bounds__(256) void shiftnmf_fft_kernel(
        const float* __restrict__ H, float* __restrict__ Hf) {
    __shared__ float xsh[4096];   // softplus(H) row, x[b + 64a]
    __shared__ float Cr[4096];    // DFT64 matrix cos(-2pi ac/64), [a*64+c]
    __shared__ float Ci[4096];    // DFT64 matrix sin(-2pi ac/64)
    __shared__ float Zr[4096];    // stage-1 output / twiddled, [b*64+c]
    __shared__ float Zi[4096];

    const int d    = blockIdx.x;
    const int tid  = threadIdx.x;
    const int lane = tid & 31;
    const int wave = tid >> 5;

    // Load row (with fused softplus) and build the shared DFT64 matrix.
    for (int i = tid; i < 4096; i += 256) {
        xsh[i] = softplus_f(H[d * MDIM + i]);
        int a = i >> 6, c = i & 63;
        int t = (a * c) & 63;                       // exact mod-64 reduction
        float ang = (-2.0f * PI_F / 64.0f) * (float)t;
        float s, co;
        __sincosf(ang, &s, &co);
        Cr[i] = co;
        Ci[i] = s;
    }
    __syncthreads();

    // WMMA operand lane mapping (ISA 7.12.2, 32-bit operands, wave32):
    //   A 16x4 : a.x=A[m][kb], a.y=A[m][kb+1],  m=lane&15, kb=(lane>>4)*2
    //   B 4x16 : b.x=B[kb][n], b.y=B[kb+1][n],  n=lane&15
    //   C/D    : vgpr j = D[j + 8*(lane>>4)][lane&15]
    const int mn  = lane & 15;
    const int kb  = (lane >> 4) << 1;
    const int rw0 = (lane >> 4) << 3;

    // ---- Stage 1: Y[b][c] = sum_a x[b+64a] * W64[a][c]   (real input) ----
    for (int r = 0; r < 2; ++r) {
        const int tile = wave * 2 + r;              // 16 tiles over 64x64
        const int b0 = (tile >> 2) << 4;
        const int c0 = (tile & 3) << 4;
        v8f accr = {}, acci = {};
        for (int a0 = 0; a0 < 64; a0 += 4) {
            v2f A, Br, Bi;
            A.x  = xsh[(b0 + mn) + 64 * (a0 + kb)];
            A.y  = xsh[(b0 + mn) + 64 * (a0 + kb + 1)];
            Br.x = Cr[(a0 + kb) * 64 + (c0 + mn)];
            Br.y = Cr[(a0 + kb + 1) * 64 + (c0 + mn)];
            Bi.x = Ci[(a0 + kb) * 64 + (c0 + mn)];
            Bi.y = Ci[(a0 + kb + 1) * 64 + (c0 + mn)];
            accr = __builtin_amdgcn_wmma_f32_16x16x4_f32(
                       false, A, false, Br, (short)0, accr, false, false);
            acci = __builtin_amdgcn_wmma_f32_16x16x4_f32(
                       false, A, false, Bi, (short)0, acci, false, false);
        }
#pragma unroll
        for (int j = 0; j < 8; ++j) {
            Zr[(b0 + rw0 + j) * 64 + (c0 + mn)] = accr[j];
            Zi[(b0 + rw0 + j) * 64 + (c0 + mn)] = acci[j];
        }
    }
    __syncthreads();

    // ---- Twiddle: z[b][c] *= exp(-2pi i * b*c / 4096) ----
    for (int i = tid; i < 4096; i += 256) {
        int b = i >> 6, c = i & 63;
        float ang = (-2.0f * PI_F / 4096.0f) * (float)(b * c);
        float s, co;
        __sincosf(ang, &s, &co);
        float yr = Zr[i], yi = Zi[i];
        Zr[i] = yr * co - yi * s;
        Zi[i] = yr * s + yi * co;
    }
    __syncthreads();

    // ---- Stage 2: Out[e][c] = sum_b W64[e][b] * z[b][c], m = c + 64e ----
    for (int r = 0; r < 2; ++r) {
        const int tile = wave * 2 + r;
        const int e0 = (tile >> 2) << 4;
        const int c0 = (tile & 3) << 4;
        v8f aRR = {}, aII = {}, aRI = {}, aIR = {};
        for (int b0 = 0; b0 < 64; b0 += 4) {
            v2f Ar, Ai, Bzr, Bzi;
            Ar.x  = Cr[(e0 + mn) * 64 + (b0 + kb)];
            Ar.y  = Cr[(e0 + mn) * 64 + (b0 + kb + 1)];
            Ai.x  = Ci[(e0 + mn) * 64 + (b0 + kb)];
            Ai.y  = Ci[(e0 + mn) * 64 + (b0 + kb + 1)];
            Bzr.x = Zr[(b0 + kb) * 64 + (c0 + mn)];
            Bzr.y = Zr[(b0 + kb + 1) * 64 + (c0 + mn)];
            Bzi.x = Zi[(b0 + kb) * 64 + (c0 + mn)];
            Bzi.y = Zi[(b0 + kb + 1) * 64 + (c0 + mn)];
            aRR = __builtin_amdgcn_wmma_f32_16x16x4_f32(
                      false, Ar, false, Bzr, (short)0, aRR, false, false);
            aII = __builtin_amdgcn_wmma_f32_16x16x4_f32(
                      false, Ai, false, Bzi, (short)0, aII, false, false);
            aRI = __builtin_amdgcn_wmma_f32_16x16x4_f32(
                      false, Ar, false, Bzi, (short)0, aRI, false, false);
            aIR = __builtin_amdgcn_wmma_f32_16x16x4_f32(
                      false, Ai, false, Bzr, (short)0, aIR, false, false);
        }
        float2* Hf2 = reinterpret_cast<float2*>(Hf);
#pragma unroll
        for (int j = 0; j < 8; ++j) {
            int e = e0 + rw0 + j;
            int m = (c0 + mn) + 64 * e;
            float2 o;
            o.x = aRR[j] - aII[j];   // Re = Cr*Zr - Ci*Zi
            o.y = aRI[j] + aIR[j];   // Im = Cr*Zi + Ci*Zr
            Hf2[d * MDIM + m] = o;
        }
    }
}

// ---------------------------------------------------------------------------
// Kernel 2: V[n,m] = sum_d spW[n,d] * exp(-2pi i tau[n,d] m/M) * Hf[d,m].
// One wave per (n, 256-wide m tile); lanes stride m by 32, 8 complex accs.
// tau/W loads are wave-uniform (-> scalar loads). Incremental rotation:
// 2 sincos per (lane, d), 7 complex-rotate steps.
// ---------------------------------------------------------------------------
__global__ __launch_bounds__(256) void shiftnmf_mod_kernel(
        const float* __restrict__ W, const float* __restrict__ tau,
        const float* __restrict__ Hf, float* __restrict__ out) {
    const int tid   = threadIdx.x;
    const int lane  = tid & 31;
    const int wid   = blockIdx.x * 8 + (tid >> 5);
    const int n     = wid >> 4;            // 16 m-tiles per row n
    const int mt    = wid & 15;
    const int mbase = mt * 256 + lane;     // this lane's first m

    float2 acc[8];
#pragma unroll
    for (int k = 0; k < 8; ++k) acc[k] = make_float2(0.0f, 0.0f);

    const float2* Hf2 = reinterpret_cast<const float2*>(Hf);

    for (int d = 0; d < RANKD; ++d) {
        float t   = tau[n * RANKD + d];            // wave-uniform -> s_load
        float spw = softplus_f(W[n * RANKD + d]);  // wave-uniform -> s_load
        // omega0 = exp(-2pi i * t * mbase / M); |theta| <= 2pi, no range issue
        float th0 = (-2.0f * PI_F / (float)MDIM) * t * (float)mbase;
        float ths = (-2.0f * PI_F / 128.0f) * t;   // step for m += 32
        float wr, wi, cs, ss;
        __sincosf(th0, &wi, &wr);
        __sincosf(ths, &ss, &cs);
#pragma unroll
        for (int k = 0; k < 8; ++k) {
            float2 h = Hf2[d * MDIM + mbase + 32 * k];
            acc[k].x += spw * (wr * h.x - wi * h.y);
            acc[k].y += spw * (wr * h.y + wi * h.x);
            float nwr = wr * cs - wi * ss;         // rotate omega by step
            wi = wr * ss + wi * cs;
            wr = nwr;
        }
    }

    float2* out2 = reinterpret_cast<float2*>(out);
#pragma unroll
    for (int k = 0; k < 8; ++k)
        out2[n * MDIM + mbase + 32 * k] = acc[k];
}

// ---------------------------------------------------------------------------
extern "C" void kernel_launch(void* const* d_in, const int* in_sizes, int n_in,
                              void* d_out, int out_size, void* d_ws, size_t ws_size,
                              hipStream_t stream) {
    const float* W   = (const float*)d_in[0];   // (1024, 32)
    const float* H   = (const float*)d_in[1];   // (32, 4096)
    const float* tau = (const float*)d_in[2];   // (1024, 32)
    float* Hf = (float*)d_ws;                   // (32, 4096) complex interleaved, 1 MB
    float* V  = (float*)d_out;                  // (1024, 4096) complex interleaved

    (void)in_sizes; (void)n_in; (void)out_size; (void)ws_size;

    shiftnmf_fft_kernel<<<RANKD, 256, 0, stream>>>(H, Hf);

    // 1024 n * 16 m-tiles = 16384 waves / 8 waves-per-block = 2048 blocks
    shiftnmf_mod_kernel<<<(NDIM * 16) / 8, 256, 0, stream>>>(W, tau, Hf, V);
}